// BalancedBCEWithLogitsLoss_11312943858133
// MI455X (gfx1250) — compile-verified
//
#include <hip/hip_runtime.h>
#include <stdint.h>

// Balanced BCE-with-logits, exact reproduction of the JAX reference:
//  - Threefry-2x32 (key = (0,42)) reproduces jax.random.uniform bit-exactly in rank order
//  - two-level radix select finds the exact K-th largest priority among negatives
//  - ties at the threshold resolved by ascending index (stable argsort semantics)
//  - loss pass reduces with V_WMMA_F32_16X16X4_F32 (ones-matrix row-sum trick)

#define N_TOTAL   16777216u
#define N_HALF    8388608u
#define TPB       256
#define BLOCKS    1024
#define COARSE_BINS 4096
#define FINE_BINS   2048
#define TIE_CAP     4096
#define FLOOR_NEG   838860u   // int(16777216 * 0.05)

// workspace layout (uint32 slots)
#define WS_NUM_POS   0
#define WS_K         1
#define WS_BSTAR     2
#define WS_CUMABOVE  3
#define WS_MSTAR     4
#define WS_T         5
#define WS_TIECNT    6
#define WS_CUTOFF    7
#define WS_HIST12    16
#define WS_HIST11    (WS_HIST12 + COARSE_BINS)
#define WS_TIES      (WS_HIST11 + FINE_BINS)
#define WS_PARTIAL   (WS_TIES + TIE_CAP)          // float[BLOCKS]
#define WS_U32_TOTAL (WS_PARTIAL + BLOCKS)

typedef __attribute__((ext_vector_type(2))) float v2f;
typedef __attribute__((ext_vector_type(8))) float v8f;

__device__ __forceinline__ uint32_t rotl32(uint32_t x, uint32_t r) {
  return (x << r) | (x >> (32u - r));
}

// Threefry-2x32, key = (0, 42)  (jax.random.key(42))
__device__ __forceinline__ void threefry_42(uint32_t x0, uint32_t x1,
                                            uint32_t& o0, uint32_t& o1) {
  const uint32_t K0 = 0u, K1 = 42u, K2 = 0u ^ 42u ^ 0x1BD11BDAu;
#define TF_R4(a, b, c, d)                         \
  x0 += x1; x1 = rotl32(x1, a); x1 ^= x0;         \
  x0 += x1; x1 = rotl32(x1, b); x1 ^= x0;         \
  x0 += x1; x1 = rotl32(x1, c); x1 ^= x0;         \
  x0 += x1; x1 = rotl32(x1, d); x1 ^= x0;
  x0 += K0; x1 += K1;
  TF_R4(13u, 15u, 26u, 6u)  x0 += K1; x1 += K2 + 1u;
  TF_R4(17u, 29u, 16u, 24u) x0 += K2; x1 += K0 + 2u;
  TF_R4(13u, 15u, 26u, 6u)  x0 += K0; x1 += K1 + 3u;
  TF_R4(17u, 29u, 16u, 24u) x0 += K1; x1 += K2 + 4u;
  TF_R4(13u, 15u, 26u, 6u)  x0 += K2; x1 += K0 + 5u;
#undef TF_R4
  o0 = x0; o1 = x1;
}

__device__ __forceinline__ float bce(float x, float y) {
  return fmaxf(x, 0.0f) - x * y + log1pf(expf(-fabsf(x)));
}

// Wave32 sum via WMMA: A = {v,0} (16x4), B = ones (4x16) -> D[m][n] = rowsum[m].
// Per-lane sum of the 8 C VGPRs gives sum(rowsum[0..7]) in lanes 0-15 and
// sum(rowsum[8..15]) in lanes 16-31; one xor-16 shuffle completes the total.
__device__ __forceinline__ float wave_sum_wmma(float v) {
  v2f a; a.x = v;    a.y = 0.0f;
  v2f b; b.x = 1.0f; b.y = 1.0f;
  v8f c = {};
  c = __builtin_amdgcn_wmma_f32_16x16x4_f32(false, a, false, b, (short)0, c,
                                            false, false);
  float s = c[0] + c[1] + c[2] + c[3] + c[4] + c[5] + c[6] + c[7];
  s += __shfl_xor(s, 16, 32);
  return s;  // total in every lane
}

__global__ void k_zero(uint32_t* ws) {
  for (uint32_t i = blockIdx.x * blockDim.x + threadIdx.x; i < WS_U32_TOTAL;
       i += gridDim.x * blockDim.x)
    ws[i] = 0u;
}

// Pass A: count positives + coarse 4096-bin histogram of m>>11 among negatives.
__global__ void k_count_hist(const float* __restrict__ label,
                             uint32_t* __restrict__ ws) {
  __shared__ uint32_t lhist[COARSE_BINS];
  __shared__ uint32_t lpos;
  for (int i = threadIdx.x; i < COARSE_BINS; i += TPB) lhist[i] = 0u;
  if (threadIdx.x == 0) lpos = 0u;
  __syncthreads();

  uint32_t mypos = 0u;
  const uint32_t stride = BLOCKS * TPB * 4u;
  for (uint32_t base = (blockIdx.x * TPB + threadIdx.x) * 4u; base < N_HALF;
       base += stride) {
    if (base + stride < N_HALF)  // in-bounds only: WGP prefetch is non-speculative
      __builtin_prefetch(label + base + stride, 0, 3);
    float4 y0 = *(const float4*)(label + base);
    float4 y1 = *(const float4*)(label + base + N_HALF);
#pragma unroll
    for (int t = 0; t < 4; ++t) {
      uint32_t b0, b1;
      threefry_42(base + t, base + t + N_HALF, b0, b1);
      float ya = (&y0.x)[t], yb = (&y1.x)[t];
      if (ya != 0.0f) mypos++; else atomicAdd(&lhist[b0 >> 20], 1u);
      if (yb != 0.0f) mypos++; else atomicAdd(&lhist[b1 >> 20], 1u);
    }
  }
  atomicAdd(&lpos, mypos);
  __syncthreads();
  for (int i = threadIdx.x; i < COARSE_BINS; i += TPB)
    if (lhist[i]) atomicAdd(&ws[WS_HIST12 + i], lhist[i]);
  if (threadIdx.x == 0) atomicAdd(&ws[WS_NUM_POS], lpos);
}

__global__ void k_select_coarse(uint32_t* ws) {
  uint32_t np = ws[WS_NUM_POS];
  uint32_t K = 3u * np;
  if (K < FLOOR_NEG) K = FLOOR_NEG;
  ws[WS_K] = K;
  uint32_t cum = 0u, bstar = 0u, cumab = 0u;
  for (int b = COARSE_BINS - 1; b >= 0; --b) {
    uint32_t h = ws[WS_HIST12 + b];
    if (cum + h >= K) { bstar = (uint32_t)b; cumab = cum; break; }
    cum += h;
  }
  ws[WS_BSTAR] = bstar;
  ws[WS_CUMABOVE] = cumab;
}

// Pass B: refine within the critical coarse bin (low 11 bits of m).
__global__ void k_fine_hist(const float* __restrict__ label,
                            uint32_t* __restrict__ ws) {
  const uint32_t bstar = ws[WS_BSTAR];
  const uint32_t stride = BLOCKS * TPB * 4u;
  for (uint32_t base = (blockIdx.x * TPB + threadIdx.x) * 4u; base < N_HALF;
       base += stride) {
    float4 y0 = *(const float4*)(label + base);
    float4 y1 = *(const float4*)(label + base + N_HALF);
#pragma unroll
    for (int t = 0; t < 4; ++t) {
      uint32_t b0, b1;
      threefry_42(base + t, base + t + N_HALF, b0, b1);
      if ((&y0.x)[t] == 0.0f && (b0 >> 20) == bstar)
        atomicAdd(&ws[WS_HIST11 + ((b0 >> 9) & 0x7FFu)], 1u);
      if ((&y1.x)[t] == 0.0f && (b1 >> 20) == bstar)
        atomicAdd(&ws[WS_HIST11 + ((b1 >> 9) & 0x7FFu)], 1u);
    }
  }
}

__global__ void k_select_fine(uint32_t* ws) {
  const uint32_t K = ws[WS_K], bstar = ws[WS_BSTAR];
  uint32_t cum = ws[WS_CUMABOVE];
  uint32_t mstar = bstar << 11, T = 0u;
  for (int f = FINE_BINS - 1; f >= 0; --f) {
    uint32_t h = ws[WS_HIST11 + f];
    if (cum + h >= K) { mstar = (bstar << 11) | (uint32_t)f; T = K - cum; break; }
    cum += h;
  }
  ws[WS_MSTAR] = mstar;
  ws[WS_T] = T;
}

// Pass C: gather indices of negatives exactly at the threshold priority.
__global__ void k_ties(const float* __restrict__ label,
                       uint32_t* __restrict__ ws) {
  const uint32_t mstar = ws[WS_MSTAR];
  const uint32_t stride = BLOCKS * TPB * 4u;
  for (uint32_t base = (blockIdx.x * TPB + threadIdx.x) * 4u; base < N_HALF;
       base += stride) {
    float4 y0 = *(const float4*)(label + base);
    float4 y1 = *(const float4*)(label + base + N_HALF);
#pragma unroll
    for (int t = 0; t < 4; ++t) {
      uint32_t b0, b1;
      threefry_42(base + t, base + t + N_HALF, b0, b1);
      if ((&y0.x)[t] == 0.0f && (b0 >> 9) == mstar) {
        uint32_t p = atomicAdd(&ws[WS_TIECNT], 1u);
        if (p < TIE_CAP) ws[WS_TIES + p] = base + t;
      }
      if ((&y1.x)[t] == 0.0f && (b1 >> 9) == mstar) {
        uint32_t p = atomicAdd(&ws[WS_TIECNT], 1u);
        if (p < TIE_CAP) ws[WS_TIES + p] = base + t + N_HALF;
      }
    }
  }
}

// Rank-select the T-th smallest tie index (stable-argsort tie break).
__global__ void k_cutoff(uint32_t* ws) {
  uint32_t tc = ws[WS_TIECNT];
  if (tc > TIE_CAP) tc = TIE_CAP;
  const uint32_t T = ws[WS_T];
  __shared__ uint32_t cut;
  if (threadIdx.x == 0) cut = 0xFFFFFFFFu;
  __syncthreads();
  if (T < tc) {
    for (uint32_t i = threadIdx.x; i < tc; i += blockDim.x) {
      uint32_t v = ws[WS_TIES + i], r = 0u;
      for (uint32_t j = 0; j < tc; ++j) r += (ws[WS_TIES + j] < v) ? 1u : 0u;
      if (r == T) cut = v;  // unique (indices are distinct)
    }
  }
  __syncthreads();
  if (threadIdx.x == 0) ws[WS_CUTOFF] = cut;
}

// Pass D: masked BCE sum; wave reduction via WMMA; fixed-slot block partials.
__global__ void k_loss(const float* __restrict__ pred,
                       const float* __restrict__ label,
                       const uint32_t* __restrict__ ws,
                       float* __restrict__ partial) {
  const uint32_t mstar = ws[WS_MSTAR];
  const uint32_t cutoff = ws[WS_CUTOFF];
  float acc = 0.0f;
  const uint32_t stride = BLOCKS * TPB * 4u;
  for (uint32_t base = (blockIdx.x * TPB + threadIdx.x) * 4u; base < N_HALF;
       base += stride) {
    if (base + stride < N_HALF) {  // in-bounds only: WGP prefetch is non-speculative
      __builtin_prefetch(pred + base + stride, 0, 3);
      __builtin_prefetch(pred + base + stride + N_HALF, 0, 3);
    }
    float4 x0 = *(const float4*)(pred + base);
    float4 x1 = *(const float4*)(pred + base + N_HALF);
    float4 y0 = *(const float4*)(label + base);
    float4 y1 = *(const float4*)(label + base + N_HALF);
#pragma unroll
    for (int t = 0; t < 4; ++t) {
      uint32_t b0, b1;
      threefry_42(base + t, base + t + N_HALF, b0, b1);
      const uint32_t m0 = b0 >> 9, m1 = b1 >> 9;
      float ya = (&y0.x)[t], yb = (&y1.x)[t];
      bool in0 = (ya != 0.0f) ||
                 (m0 > mstar || (m0 == mstar && (base + t) < cutoff));
      bool in1 = (yb != 0.0f) ||
                 (m1 > mstar || (m1 == mstar && (base + t + N_HALF) < cutoff));
      if (in0) acc += bce((&x0.x)[t], ya);
      if (in1) acc += bce((&x1.x)[t], yb);
    }
  }
  // all 32 lanes convergent here (EXEC all-ones for WMMA)
  float tot = wave_sum_wmma(acc);
  __shared__ float wsum[TPB / 32];
  const int lane = threadIdx.x & 31, wid = threadIdx.x >> 5;
  if (lane == 0) wsum[wid] = tot;
  __syncthreads();
  if (threadIdx.x == 0) {
    float s = 0.0f;
#pragma unroll
    for (int w = 0; w < TPB / 32; ++w) s += wsum[w];
    partial[blockIdx.x] = s;
  }
}

__global__ void k_final(const uint32_t* __restrict__ ws,
                        const float* __restrict__ partial,
                        float* __restrict__ out) {
  __shared__ float red[TPB];
  float s = 0.0f;
  for (uint32_t i = threadIdx.x; i < BLOCKS; i += TPB) s += partial[i];
  red[threadIdx.x] = s;
  __syncthreads();
  for (int off = TPB / 2; off > 0; off >>= 1) {
    if ((int)threadIdx.x < off) red[threadIdx.x] += red[threadIdx.x + off];
    __syncthreads();
  }
  if (threadIdx.x == 0) {
    // |selected| = num_pos + K exactly (K negatives always available)
    float denom = (float)(ws[WS_NUM_POS] + ws[WS_K]);
    out[0] = 1.0f * (red[0] / denom);
  }
}

extern "C" void kernel_launch(void* const* d_in, const int* in_sizes, int n_in,
                              void* d_out, int out_size, void* d_ws,
                              size_t ws_size, hipStream_t stream) {
  (void)in_sizes; (void)n_in; (void)out_size; (void)ws_size;
  const float* pred = (const float*)d_in[0];
  const float* label = (const float*)d_in[1];
  uint32_t* ws = (uint32_t*)d_ws;
  float* out = (float*)d_out;
  float* partial = (float*)(ws + WS_PARTIAL);

  k_zero<<<64, TPB, 0, stream>>>(ws);
  k_count_hist<<<BLOCKS, TPB, 0, stream>>>(label, ws);
  k_select_coarse<<<1, 1, 0, stream>>>(ws);
  k_fine_hist<<<BLOCKS, TPB, 0, stream>>>(label, ws);
  k_select_fine<<<1, 1, 0, stream>>>(ws);
  k_ties<<<BLOCKS, TPB, 0, stream>>>(label, ws);
  k_cutoff<<<1, TPB, 0, stream>>>(ws);
  k_loss<<<BLOCKS, TPB, 0, stream>>>(pred, label, ws, partial);
  k_final<<<1, TPB, 0, stream>>>(ws, partial, out);
}